// NaiveEuclideanGNN_66348654788628
// MI455X (gfx1250) — compile-verified
//
#include <hip/hip_runtime.h>
#include <hip/hip_bf16.h>
#include <math.h>

#define N_NODES 50000
#define N_EDGES 800000
#define N_GRAPHS 512
#define HID 128
#define N_LAYERS 4

typedef __attribute__((ext_vector_type(2))) float v2f;
typedef __attribute__((ext_vector_type(8))) float v8f;

// ---------------------------------------------------------------------------
// WMMA fp32 GEMM: out[M,128] = act( (A (+Aadd)) [M,K] @ W [K,128] + b )
// One wave computes a 32x128 tile (two 16-row tiles) with V_WMMA_F32_16X16X4_F32
// so each B (weight) fragment load is reused by two independent WMMAs.
// A-frag (16x4): lane(0..15)=rows, v0 = K0|K2 halves, v1 = K1|K3.
// B-frag (4x16): lane(0..15)=cols, same K striping.
// C/D (16x16):   acc[r] holds row r (lanes 0-15) / row r+8 (lanes 16-31).
// M must be a multiple of 16; K a multiple of 4. If M%32==16, the last wave
// clamps its second tile to a duplicate of the first (wave-uniform, EXEC=all 1s)
// and skips the duplicate store.
// ---------------------------------------------------------------------------
template<bool HAS_ADD, bool RELU>
__global__ __launch_bounds__(256, 1) void wmma_gemm_kernel(
        const float* __restrict__ A,
        const float* __restrict__ Aadd,   // used iff HAS_ADD
        const float* __restrict__ W,      // [K,128]
        const float* __restrict__ bias,   // [128]
        float* __restrict__ out,          // [M,128]
        int M, int K)
{
    const int wave = (blockIdx.x * blockDim.x + threadIdx.x) >> 5;
    const int row0 = wave * 32;
    if (row0 >= M) return;                       // whole-wave uniform

    const bool has2 = (row0 + 16) < M;           // wave-uniform
    const int  row1 = has2 ? (row0 + 16) : row0; // clamp: duplicate compute

    const int lane = threadIdx.x & 31;
    const int lrow = lane & 15;                  // row (A) / col (B) in tile
    const int kh   = lane >> 4;                  // K-half select

    const float* __restrict__ arow0 = A + (size_t)(row0 + lrow) * K;
    const float* __restrict__ arow1 = A + (size_t)(row1 + lrow) * K;
    const float* __restrict__ brow0 = HAS_ADD ? (Aadd + (size_t)(row0 + lrow) * K) : nullptr;
    const float* __restrict__ brow1 = HAS_ADD ? (Aadd + (size_t)(row1 + lrow) * K) : nullptr;

    v8f acc0[8], acc1[8];
#pragma unroll
    for (int t = 0; t < 8; ++t) {
        acc0[t] = (v8f){0.f,0.f,0.f,0.f,0.f,0.f,0.f,0.f};
        acc1[t] = (v8f){0.f,0.f,0.f,0.f,0.f,0.f,0.f,0.f};
    }

    for (int k = 0; k < K; k += 4) {
        const int ka = k + 2 * kh;
        v2f a0, a1;
        a0.x = arow0[ka];
        a0.y = arow0[ka + 1];
        a1.x = arow1[ka];
        a1.y = arow1[ka + 1];
        if (HAS_ADD) {
            a0.x += brow0[ka];
            a0.y += brow0[ka + 1];
            a1.x += brow1[ka];
            a1.y += brow1[ka + 1];
        }

        const float* __restrict__ w0 = W + (size_t)ka * HID + lrow;
#pragma unroll
        for (int t = 0; t < 8; ++t) {
            v2f b;
            b.x = w0[t * 16];              // W[ka    ][t*16 + lrow]
            b.y = w0[HID + t * 16];        // W[ka + 1][t*16 + lrow]
            acc0[t] = __builtin_amdgcn_wmma_f32_16x16x4_f32(
                false, a0, false, b, (short)0, acc0[t], false, false);
            acc1[t] = __builtin_amdgcn_wmma_f32_16x16x4_f32(
                false, a1, false, b, (short)0, acc1[t], false, false);
        }
    }

#pragma unroll
    for (int t = 0; t < 8; ++t) {
        const int col = t * 16 + lrow;
        const float bv = bias[col];
#pragma unroll
        for (int r = 0; r < 8; ++r) {
            float v = acc0[t][r] + bv;
            if (RELU) v = fmaxf(v, 0.f);
            out[(size_t)(row0 + r + 8 * kh) * HID + col] = v;
        }
    }
    if (has2) {
#pragma unroll
        for (int t = 0; t < 8; ++t) {
            const int col = t * 16 + lrow;
            const float bv = bias[col];
#pragma unroll
            for (int r = 0; r < 8; ++r) {
                float v = acc1[t][r] + bv;
                if (RELU) v = fmaxf(v, 0.f);
                out[(size_t)(row1 + r + 8 * kh) * HID + col] = v;
            }
        }
    }
}

// ---------------------------------------------------------------------------
// pre[N,256] = [ embed_tab[z[n]] , pos[n] @ W_pos + b_pos ]
// ---------------------------------------------------------------------------
__global__ void pre_embed_kernel(const int* __restrict__ z,
                                 const float* __restrict__ pos,
                                 const float* __restrict__ embed_tab,
                                 const float* __restrict__ W_pos,   // [3,128]
                                 const float* __restrict__ b_pos,   // [128]
                                 float* __restrict__ pre, int n)
{
    const int idx = blockIdx.x * blockDim.x + threadIdx.x;
    if (idx >= n * HID) return;
    const int node = idx >> 7;
    const int f    = idx & (HID - 1);
    const float px = pos[node * 3 + 0];
    const float py = pos[node * 3 + 1];
    const float pz = pos[node * 3 + 2];
    pre[(size_t)node * 256 + f] = embed_tab[(size_t)z[node] * HID + f];
    pre[(size_t)node * 256 + HID + f] =
        px * W_pos[f] + py * W_pos[HID + f] + pz * W_pos[2 * HID + f] + b_pos[f];
}

// ---------------------------------------------------------------------------
// agg[dst] += x[src] : one wave per edge, float4 per lane, fp32 global atomics
// ---------------------------------------------------------------------------
__global__ void scatter_add_kernel(const float* __restrict__ x,
                                   const int* __restrict__ src,
                                   const int* __restrict__ dst,
                                   float* __restrict__ agg, int nEdges)
{
    const int idx = blockIdx.x * blockDim.x + threadIdx.x;
    const int e = idx >> 5;
    if (e >= nEdges) return;
    const int lane = idx & 31;
    const int s = src[e];
    const int d = dst[e];
    const float4 v = *(const float4*)(x + (size_t)s * HID + lane * 4);
    float* p = agg + (size_t)d * HID + lane * 4;
    atomicAdd(p + 0, v.x);
    atomicAdd(p + 1, v.y);
    atomicAdd(p + 2, v.z);
    atomicAdd(p + 3, v.w);
}

__global__ void zero_kernel(float* __restrict__ p, size_t n)
{
    const size_t idx = (size_t)blockIdx.x * blockDim.x + threadIdx.x;
    if (idx < n) p[idx] = 0.f;
}

// ---------------------------------------------------------------------------
// pooled[batch[n]] += x[n]
// ---------------------------------------------------------------------------
__global__ void pool_kernel(const float* __restrict__ x,
                            const int* __restrict__ batch,
                            float* __restrict__ pooled, int n)
{
    const int idx = blockIdx.x * blockDim.x + threadIdx.x;
    if (idx >= n * HID) return;
    const int node = idx >> 7;
    const int f    = idx & (HID - 1);
    atomicAdd(&pooled[(size_t)batch[node] * HID + f], x[(size_t)node * HID + f]);
}

// ---------------------------------------------------------------------------
// Final heads: raw[k] = hh[k][g] . head_W2[k] + head_b2[k]; evidential math.
// One block per graph, one wave per head (wave32 shuffle reduction).
// Output order: gamma, aleatoric, epistemic, nu, alpha, beta (each [512]).
// ---------------------------------------------------------------------------
__device__ inline float softplus_f(float v)
{
    return fmaxf(v, 0.f) + log1pf(expf(-fabsf(v)));
}

__global__ void head_reduce_kernel(const float* __restrict__ hh,   // [4,512,128]
                                   const float* __restrict__ W2,   // [4,128]
                                   const float* __restrict__ b2,   // [4]
                                   float* __restrict__ out)        // [6,512]
{
    __shared__ float raw[4];
    const int g    = blockIdx.x;
    const int k    = threadIdx.x >> 5;
    const int lane = threadIdx.x & 31;

    const float* __restrict__ h = hh + ((size_t)k * N_GRAPHS + g) * HID;
    const float* __restrict__ w = W2 + (size_t)k * HID;
    float s = 0.f;
#pragma unroll
    for (int f = lane; f < HID; f += 32) s += h[f] * w[f];
#pragma unroll
    for (int off = 16; off > 0; off >>= 1) s += __shfl_down(s, off, 32);
    if (lane == 0) raw[k] = s + b2[k];
    __syncthreads();

    if (threadIdx.x == 0) {
        const float alpha = fmaxf(softplus_f(raw[0]) + 1.f, 1.f + 1e-4f);
        const float beta  = softplus_f(raw[1]);
        const float nu    = softplus_f(raw[2]);
        const float gamma = raw[3];
        const float am1   = alpha - 1.f;
        out[0 * N_GRAPHS + g] = gamma;              // gamma
        out[1 * N_GRAPHS + g] = beta / am1;         // aleatoric
        out[2 * N_GRAPHS + g] = beta / (am1 * nu);  // epistemic
        out[3 * N_GRAPHS + g] = nu;
        out[4 * N_GRAPHS + g] = alpha;
        out[5 * N_GRAPHS + g] = beta;
    }
}

// ---------------------------------------------------------------------------

static inline void launch_gemm(const float* A, const float* Aadd, const float* W,
                               const float* bias, float* out, int M, int K,
                               bool relu, hipStream_t stream)
{
    const int waves   = (M + 31) / 32;       // 32 rows per wave
    const int threads = waves * 32;
    const int block   = 256;
    const int grid    = (threads + block - 1) / block;
    if (Aadd) {
        if (relu) wmma_gemm_kernel<true, true ><<<grid, block, 0, stream>>>(A, Aadd, W, bias, out, M, K);
        else      wmma_gemm_kernel<true, false><<<grid, block, 0, stream>>>(A, Aadd, W, bias, out, M, K);
    } else {
        if (relu) wmma_gemm_kernel<false, true ><<<grid, block, 0, stream>>>(A, Aadd, W, bias, out, M, K);
        else      wmma_gemm_kernel<false, false><<<grid, block, 0, stream>>>(A, Aadd, W, bias, out, M, K);
    }
}

extern "C" void kernel_launch(void* const* d_in, const int* in_sizes, int n_in,
                              void* d_out, int out_size, void* d_ws, size_t ws_size,
                              hipStream_t stream)
{
    const int*   z          = (const int*)  d_in[0];
    const float* pos        = (const float*)d_in[1];
    const int*   batch      = (const int*)  d_in[2];
    const int*   edge_index = (const int*)  d_in[3];
    const float* embed_tab  = (const float*)d_in[4];
    const float* W_pos      = (const float*)d_in[5];
    const float* b_pos      = (const float*)d_in[6];
    const float* W_comb     = (const float*)d_in[7];
    const float* b_comb     = (const float*)d_in[8];
    const float* gin_W1     = (const float*)d_in[9];
    const float* gin_b1     = (const float*)d_in[10];
    const float* gin_W2     = (const float*)d_in[11];
    const float* gin_b2     = (const float*)d_in[12];
    const float* head_W1    = (const float*)d_in[13];
    const float* head_b1    = (const float*)d_in[14];
    const float* head_W2    = (const float*)d_in[15];
    const float* head_b2    = (const float*)d_in[16];
    float* out = (float*)d_out;

    const int* src = edge_index;             // row 0
    const int* dst = edge_index + N_EDGES;   // row 1

    // Workspace layout (floats):
    //   x[N,128] | agg[N,128] | tmp[N,128] | pooled[512,128] | hh[4,512,128]
    // pre[N,256] aliases agg+tmp (only live before the GIN stack starts).
    float* x      = (float*)d_ws;
    float* agg    = x   + (size_t)N_NODES * HID;
    float* tmp    = agg + (size_t)N_NODES * HID;
    float* pre    = agg;                                  // N*256 floats
    float* pooled = tmp + (size_t)N_NODES * HID;
    float* hh     = pooled + (size_t)N_GRAPHS * HID;

    const int nodeThreads = N_NODES * HID;
    const int nodeBlocks  = (nodeThreads + 255) / 256;

    // 1) Initial embedding: pre = [embed(z), pos@W_pos+b_pos]; x = relu(pre@W_comb+b)
    pre_embed_kernel<<<nodeBlocks, 256, 0, stream>>>(z, pos, embed_tab, W_pos,
                                                     b_pos, pre, N_NODES);
    launch_gemm(pre, nullptr, W_comb, b_comb, x, N_NODES, 2 * HID, true, stream);

    // 2) GIN stack
    const int edgeThreads = N_EDGES * 32;
    const int edgeBlocks  = (edgeThreads + 255) / 256;
    for (int l = 0; l < N_LAYERS; ++l) {
        zero_kernel<<<nodeBlocks, 256, 0, stream>>>(agg, (size_t)N_NODES * HID);
        scatter_add_kernel<<<edgeBlocks, 256, 0, stream>>>(x, src, dst, agg, N_EDGES);
        // tmp = relu((x + agg) @ W1 + b1)
        launch_gemm(x, agg, gin_W1 + (size_t)l * HID * HID,
                    gin_b1 + (size_t)l * HID, tmp, N_NODES, HID, true, stream);
        // x = (tmp @ W2 + b2), relu between layers but not after the last
        launch_gemm(tmp, nullptr, gin_W2 + (size_t)l * HID * HID,
                    gin_b2 + (size_t)l * HID, x, N_NODES, HID,
                    l < N_LAYERS - 1, stream);
    }

    // 3) Sum pooling per graph
    const size_t poolN = (size_t)N_GRAPHS * HID;
    zero_kernel<<<(int)((poolN + 255) / 256), 256, 0, stream>>>(pooled, poolN);
    pool_kernel<<<nodeBlocks, 256, 0, stream>>>(x, batch, pooled, N_NODES);

    // 4) Head hidden layers: hh[k] = relu(pooled @ head_W1[k] + head_b1[k])
    for (int k = 0; k < 4; ++k) {
        launch_gemm(pooled, nullptr, head_W1 + (size_t)k * HID * HID,
                    head_b1 + (size_t)k * HID, hh + (size_t)k * N_GRAPHS * HID,
                    N_GRAPHS, HID, true, stream);
    }

    // 5) Head outputs + evidential math
    head_reduce_kernel<<<N_GRAPHS, 128, 0, stream>>>(hh, head_W2, head_b2, out);
}